// LSPConditionalGNN_56100862820768
// MI455X (gfx1250) — compile-verified
//
#include <hip/hip_runtime.h>

typedef float v8f __attribute__((ext_vector_type(8)));
typedef float v2f __attribute__((ext_vector_type(2)));

#define SLOPE 0.1f
#define BN_EPS 1e-5f

// ---------------------------------------------------------------------------
// Degree / inverse-count
// ---------------------------------------------------------------------------
__global__ void __launch_bounds__(256)
degree_kernel(const int* __restrict__ e0, const int* __restrict__ e1, int E,
              float* __restrict__ cnt) {
  int i = blockIdx.x * blockDim.x + threadIdx.x;
  if (i < E) {
    atomicAdd(&cnt[e1[i]], 1.0f);  // dst side of (src=e0, dst=e1)
    atomicAdd(&cnt[e0[i]], 1.0f);  // dst side of reversed edge
  }
}

__global__ void __launch_bounds__(256)
inv_kernel(float* __restrict__ cnt, int n) {
  int i = blockIdx.x * blockDim.x + threadIdx.x;
  if (i < n) cnt[i] = 1.0f / fmaxf(cnt[i], 1.0f);
}

// ---------------------------------------------------------------------------
// Edge scatter: AGG[dst, c] += H[src, c] for both directions, 8 channels
// ---------------------------------------------------------------------------
__global__ void __launch_bounds__(256)
scatter_kernel(const int* __restrict__ e0, const int* __restrict__ e1, int E,
               const float* __restrict__ H, float* __restrict__ AGG) {
  int g = blockIdx.x * blockDim.x + threadIdx.x;  // 2*E*8 threads
  int total = E * 16;
  if (g >= total) return;
  int c = g & 7;
  int e = g >> 3;
  int s, d;
  if (e < E) { s = e0[e]; d = e1[e]; }
  else       { s = e1[e - E]; d = e0[e - E]; }
  atomicAdd(&AGG[d * 8 + c], H[s * 8 + c]);
}

// ---------------------------------------------------------------------------
// Branch-free v2f operand loaders. Region choice keys off loop-constant k0
// (compile-time after unroll); lane-dependent cases use value selects only.
//   MODE 0: fc1 input = concat[lf(64) | history | is_subgoal] (KPAD=68)
//   MODE 1: lrelu(scale*Y + shift) of previous layer (IN_DIM = 32 or 16)
//   MODE 2: conv input = concat[AGG*inv (8) | H (8)]
// ---------------------------------------------------------------------------
template <int MODE, int IN_DIM>
__device__ __forceinline__ v2f loadA2(const float* __restrict__ A0,
                                      const float* __restrict__ A1,
                                      const float* __restrict__ A2,
                                      const float* __restrict__ scale,
                                      const float* __restrict__ shift,
                                      int row, int k0, int khalf) {
  const int ka = k0 + 2 * khalf;
  v2f r;
  if constexpr (MODE == 0) {
    if (k0 < 64) {  // compile-time after unroll
      r.x = A0[row * 64 + ka];
      r.y = A0[row * 64 + ka + 1];
    } else {
      // k0 == 64: khalf==0 -> (history, is_subgoal); khalf==1 -> padding
      float h = A1[row];
      float s = A2[row];
      r.x = khalf ? 0.0f : h;
      r.y = khalf ? 0.0f : s;
    }
  } else if constexpr (MODE == 1) {
    float y0 = A0[row * IN_DIM + ka];
    float y1 = A0[row * IN_DIM + ka + 1];
    float v0 = scale[ka] * y0 + shift[ka];
    float v1 = scale[ka + 1] * y1 + shift[ka + 1];
    r.x = v0 >= 0.0f ? v0 : SLOPE * v0;
    r.y = v1 >= 0.0f ? v1 : SLOPE * v1;
  } else {
    if (k0 < 8) {
      float inv = A1[row];
      r.x = A0[row * 8 + ka] * inv;
      r.y = A0[row * 8 + ka + 1] * inv;
    } else {
      r.x = A2[row * 8 + (ka - 8)];
      r.y = A2[row * 8 + (ka - 7)];
    }
  }
  return r;
}

template <int MODE, int IN_DIM>
__device__ __forceinline__ v2f loadB2(const float* __restrict__ W,
                                      const float* __restrict__ W2, int n,
                                      int k0, int khalf) {
  const int ka = k0 + 2 * khalf;
  v2f r;
  if constexpr (MODE == 2) {
    if (k0 < 8) {
      r.x = W[n * 8 + ka];
      r.y = W[n * 8 + ka + 1];
    } else {
      r.x = W2[n * 8 + (ka - 8)];
      r.y = W2[n * 8 + (ka - 7)];
    }
  } else if constexpr (IN_DIM == 66) {
    if (k0 < 64) {
      r.x = W[n * 66 + ka];
      r.y = W[n * 66 + ka + 1];
    } else {
      float w0 = W[n * 66 + 64];
      float w1 = W[n * 66 + 65];
      r.x = khalf ? 0.0f : w0;
      r.y = khalf ? 0.0f : w1;
    }
  } else {
    r.x = W[n * IN_DIM + ka];
    r.y = W[n * IN_DIM + ka + 1];
  }
  return r;
}

// ---------------------------------------------------------------------------
// Wave-level GEMM: one wave computes a 16-node x OUT_DIM tile of
//   Y = X @ W^T + bias   using V_WMMA_F32_16X16X4_F32
// A frag (16x4 f32): lane m = lane&15, half selects K pair, v2f = (K, K+1)
// B frag (4x16 f32): lane n = lane&15, same (k0 + 2*khalf) indexing
// D (16x16 f32): vgpr r -> row r + 8*(lane>=16), col = lane&15
// Row/col addresses are clamped (not guarded) so the hot loop has no
// divergence; out-of-range lanes compute garbage that is never stored.
// ---------------------------------------------------------------------------
template <int MODE, int IN_DIM, int OUT_DIM>
__global__ void __launch_bounds__(256)
gemm_wmma_kernel(const float* __restrict__ A0, const float* __restrict__ A1,
                 const float* __restrict__ A2, const float* __restrict__ scale,
                 const float* __restrict__ shift, const float* __restrict__ W,
                 const float* __restrict__ W2, const float* __restrict__ bias,
                 float* __restrict__ Y, int n_rows) {
  constexpr int KPAD = (IN_DIM + 3) & ~3;
  constexpr int NT = (OUT_DIM + 15) / 16;

  const int lane = threadIdx.x & 31;
  const int wave = threadIdx.x >> 5;
  const int waves = blockDim.x >> 5;
  const int rowTile = blockIdx.x * waves + wave;
  const int row0 = rowTile * 16;
  if (row0 >= n_rows) return;  // wave-uniform: EXEC all-ones for WMMA

  const int khalf = lane >> 4;
  const int l16 = lane & 15;
  const int arow = min(row0 + l16, n_rows - 1);  // clamp, no divergence

  v8f acc[NT];
#pragma unroll
  for (int t = 0; t < NT; ++t)
    acc[t] = v8f{0.f, 0.f, 0.f, 0.f, 0.f, 0.f, 0.f, 0.f};

#pragma unroll
  for (int k0 = 0; k0 < KPAD; k0 += 4) {
    v2f a = loadA2<MODE, IN_DIM>(A0, A1, A2, scale, shift, arow, k0, khalf);
#pragma unroll
    for (int t = 0; t < NT; ++t) {
      const int nb = min(t * 16 + l16, OUT_DIM - 1);  // clamp address
      v2f b = loadB2<MODE, IN_DIM>(W, W2, nb, k0, khalf);
      acc[t] = __builtin_amdgcn_wmma_f32_16x16x4_f32(
          false, a, false, b, (short)0, acc[t], false, false);
    }
  }

#pragma unroll
  for (int t = 0; t < NT; ++t) {
    const int ncol = t * 16 + l16;
    const float bv = bias[min(ncol, OUT_DIM - 1)];
    if (ncol < OUT_DIM) {
#pragma unroll
      for (int r = 0; r < 8; ++r) {
        const int row = row0 + r + 8 * khalf;
        if (row < n_rows) Y[row * OUT_DIM + ncol] = acc[t][r] + bv;
      }
    }
  }
}

// ---------------------------------------------------------------------------
// Per-channel sum / sum-of-squares over Y[N, C]  (C divides 256)
// ---------------------------------------------------------------------------
template <int C>
__global__ void __launch_bounds__(256)
colstats_kernel(const float* __restrict__ Y, int n, float* __restrict__ sum,
                float* __restrict__ sumsq) {
  __shared__ float s1[256];
  __shared__ float s2[256];
  const int t = threadIdx.x;
  const int c = t % C;
  const int rpb = 256 / C;
  float a = 0.0f, b = 0.0f;
  for (int r = blockIdx.x * rpb + t / C; r < n; r += gridDim.x * rpb) {
    float v = Y[r * C + c];
    a += v;
    b += v * v;
  }
  s1[t] = a;
  s2[t] = b;
  __syncthreads();
  for (int off = 128; off >= C; off >>= 1) {
    if (t < off) {
      s1[t] += s1[t + off];
      s2[t] += s2[t + off];
    }
    __syncthreads();
  }
  if (t < C) {
    atomicAdd(&sum[c], s1[t]);
    atomicAdd(&sumsq[c], s2[t]);
  }
}

// scale = gamma * rsqrt(var + eps); shift = beta - scale * mean
__global__ void bn_finalize_kernel(const float* __restrict__ sum,
                                   const float* __restrict__ sumsq,
                                   const float* __restrict__ gamma,
                                   const float* __restrict__ beta,
                                   float* __restrict__ scale,
                                   float* __restrict__ shift, int C,
                                   float invN) {
  int c = threadIdx.x;
  if (c < C) {
    float m = sum[c] * invN;
    float v = sumsq[c] * invN - m * m;
    float a = gamma[c] * rsqrtf(v + BN_EPS);
    scale[c] = a;
    shift[c] = beta[c] - a * m;
  }
}

// H = lrelu(scale*Y + shift), C = 8
__global__ void __launch_bounds__(256)
bn_apply8_kernel(const float* __restrict__ Y, const float* __restrict__ scale,
                 const float* __restrict__ shift, float* __restrict__ H,
                 int n) {
  int i = blockIdx.x * blockDim.x + threadIdx.x;
  if (i < n * 8) {
    int c = i & 7;
    float v = scale[c] * Y[i] + shift[c];
    H[i] = v >= 0.0f ? v : SLOPE * v;
  }
}

// out[n,3] = H[n,8] @ Wc^T + bc
__global__ void __launch_bounds__(256)
classifier_kernel(const float* __restrict__ H, const float* __restrict__ Wc,
                  const float* __restrict__ bc, float* __restrict__ out,
                  int n) {
  int i = blockIdx.x * blockDim.x + threadIdx.x;
  if (i < n * 3) {
    int row = i / 3, c = i - row * 3;
    float acc = bc[c];
#pragma unroll
    for (int k = 0; k < 8; ++k) acc += H[row * 8 + k] * Wc[c * 8 + k];
    out[i] = acc;
  }
}

// ---------------------------------------------------------------------------
// Host orchestration (graph-capture safe: only kernels + hipMemsetAsync)
// ---------------------------------------------------------------------------
extern "C" void kernel_launch(void* const* d_in, const int* in_sizes, int n_in,
                              void* d_out, int out_size, void* d_ws,
                              size_t ws_size, hipStream_t stream) {
  const int N = in_sizes[1];       // history is [N]
  const int E = in_sizes[3] / 2;   // edge_data is [2, E]

  const float* lf   = (const float*)d_in[0];
  const float* hist = (const float*)d_in[1];
  const float* sub  = (const float*)d_in[2];
  const int*   ed   = (const int*)d_in[3];
  // params in setup_inputs() insertion order
  const float* W1 = (const float*)d_in[4];
  const float* b1 = (const float*)d_in[5];
  const float* W2 = (const float*)d_in[6];
  const float* b2 = (const float*)d_in[7];
  const float* W3 = (const float*)d_in[8];
  const float* b3 = (const float*)d_in[9];
  const float* Wc = (const float*)d_in[10];
  const float* bc = (const float*)d_in[11];
  const float* WL[3] = {(const float*)d_in[12], (const float*)d_in[15],
                        (const float*)d_in[18]};
  const float* BL[3] = {(const float*)d_in[13], (const float*)d_in[16],
                        (const float*)d_in[19]};
  const float* WR[3] = {(const float*)d_in[14], (const float*)d_in[17],
                        (const float*)d_in[20]};
  const float* G[6] = {(const float*)d_in[21], (const float*)d_in[23],
                       (const float*)d_in[25], (const float*)d_in[27],
                       (const float*)d_in[29], (const float*)d_in[31]};
  const float* B[6] = {(const float*)d_in[22], (const float*)d_in[24],
                       (const float*)d_in[26], (const float*)d_in[28],
                       (const float*)d_in[30], (const float*)d_in[32]};

  // workspace layout (floats): ~73N + 768 total (~29.3 MB)
  float* ws   = (float*)d_ws;
  float* cnt  = ws;                        // N   (becomes inv-degree)
  float* Y32  = cnt + (size_t)N;           // 32N
  float* Y16  = Y32 + (size_t)32 * N;      // 16N
  float* Y8   = Y16 + (size_t)16 * N;      // 8N
  float* H    = Y8 + (size_t)8 * N;        // 8N
  float* AGG  = H + (size_t)8 * N;         // 8N
  float* STAT = AGG + (size_t)8 * N;       // 6 layers x (32 sum + 32 sumsq)
  float* SS   = STAT + 384;                // 6 layers x (32 scale + 32 shift)

  const int* e0 = ed;
  const int* e1 = ed + E;
  const float invN = 1.0f / (float)N;

  hipMemsetAsync(cnt, 0, (size_t)N * sizeof(float), stream);
  hipMemsetAsync(STAT, 0, 384 * sizeof(float), stream);

  degree_kernel<<<(E + 255) / 256, 256, 0, stream>>>(e0, e1, E, cnt);
  inv_kernel<<<(N + 255) / 256, 256, 0, stream>>>(cnt, N);

  const int rowTiles = (N + 15) / 16;
  const int gblocks = (rowTiles + 7) / 8;  // 8 waves (tiles) per 256-thr block

  // fc1: [N,66] -> [N,32]
  gemm_wmma_kernel<0, 66, 32><<<gblocks, 256, 0, stream>>>(
      lf, hist, sub, nullptr, nullptr, W1, nullptr, b1, Y32, N);
  colstats_kernel<32><<<512, 256, 0, stream>>>(Y32, N, STAT + 0, STAT + 32);
  bn_finalize_kernel<<<1, 32, 0, stream>>>(STAT + 0, STAT + 32, G[0], B[0],
                                           SS + 0, SS + 32, 32, invN);
  // fc2: BN+lrelu of fc1 fused into A-loader
  gemm_wmma_kernel<1, 32, 16><<<gblocks, 256, 0, stream>>>(
      Y32, nullptr, nullptr, SS + 0, SS + 32, W2, nullptr, b2, Y16, N);
  colstats_kernel<16><<<512, 256, 0, stream>>>(Y16, N, STAT + 64, STAT + 96);
  bn_finalize_kernel<<<1, 32, 0, stream>>>(STAT + 64, STAT + 96, G[1], B[1],
                                           SS + 64, SS + 96, 16, invN);
  // fc3
  gemm_wmma_kernel<1, 16, 8><<<gblocks, 256, 0, stream>>>(
      Y16, nullptr, nullptr, SS + 64, SS + 96, W3, nullptr, b3, Y8, N);
  colstats_kernel<8><<<512, 256, 0, stream>>>(Y8, N, STAT + 128, STAT + 160);
  bn_finalize_kernel<<<1, 32, 0, stream>>>(STAT + 128, STAT + 160, G[2], B[2],
                                           SS + 128, SS + 160, 8, invN);
  bn_apply8_kernel<<<(8 * N + 255) / 256, 256, 0, stream>>>(Y8, SS + 128,
                                                            SS + 160, H, N);

  // 3 SAGE conv layers
  const int sthreads = E * 16;  // 2E directed edges x 8 channels
  for (int ci = 0; ci < 3; ++ci) {
    hipMemsetAsync(AGG, 0, (size_t)8 * N * sizeof(float), stream);
    scatter_kernel<<<(sthreads + 255) / 256, 256, 0, stream>>>(e0, e1, E, H,
                                                               AGG);
    gemm_wmma_kernel<2, 16, 8><<<gblocks, 256, 0, stream>>>(
        AGG, cnt, H, nullptr, nullptr, WL[ci], WR[ci], BL[ci], Y8, N);
    const int li = 3 + ci;
    colstats_kernel<8><<<512, 256, 0, stream>>>(Y8, N, STAT + li * 64,
                                                STAT + li * 64 + 32);
    bn_finalize_kernel<<<1, 32, 0, stream>>>(
        STAT + li * 64, STAT + li * 64 + 32, G[li], B[li], SS + li * 64,
        SS + li * 64 + 32, 8, invN);
    bn_apply8_kernel<<<(8 * N + 255) / 256, 256, 0, stream>>>(
        Y8, SS + li * 64, SS + li * 64 + 32, H, N);
  }

  classifier_kernel<<<(3 * N + 255) / 256, 256, 0, stream>>>(H, Wc, bc,
                                                             (float*)d_out, N);
  (void)n_in;
  (void)out_size;
  (void)ws_size;
}